// acm_hnode_prompt_layer_feature_weighted_sum_21534966022304
// MI455X (gfx1250) — compile-verified
//
#include <hip/hip_runtime.h>

typedef long long ll;

// ---------------------------------------------------------------------------
// ELU (alpha = 1), matching jax.nn.elu in f32.
// ---------------------------------------------------------------------------
__device__ __forceinline__ float elu1(float x) {
    return x > 0.0f ? x : (__expf(x) - 1.0f);
}

// ---------------------------------------------------------------------------
// Zero the output (harness poisons d_out with 0xAA before timing).
// ---------------------------------------------------------------------------
__global__ void zero_f4(float4* __restrict__ p, ll n4) {
    ll i = blockIdx.x * (ll)blockDim.x + threadIdx.x;
    if (i < n4) p[i] = make_float4(0.f, 0.f, 0.f, 0.f);
}

__global__ void zero_f1(float* __restrict__ p, ll n) {
    ll i = blockIdx.x * (ll)blockDim.x + threadIdx.x;
    if (i < n) p[i] = 0.0f;
}

// ---------------------------------------------------------------------------
// Precompute emb = elu(G * w) into workspace (vectorized float4).
// ---------------------------------------------------------------------------
__global__ void emb_elu(const float4* __restrict__ g, const float4* __restrict__ w,
                        float4* __restrict__ emb, ll n4, int d4) {
    ll i = blockIdx.x * (ll)blockDim.x + threadIdx.x;
    if (i >= n4) return;
    int col = (int)(i % d4);
    float4 gv = g[i];
    float4 wv = w[col];
    float4 o;
    o.x = elu1(gv.x * wv.x);
    o.y = elu1(gv.y * wv.y);
    o.z = elu1(gv.z * wv.z);
    o.w = elu1(gv.w * wv.w);
    emb[i] = o;
}

// ---------------------------------------------------------------------------
// CDNA5 async global->LDS copy (gfx1250). Signature learned from the compiler
// diagnostic: (v4i __device__* gsrc, v4i __shared__* ldst, imm, imm).
// Guarded so absence falls back to direct global loads + global_prefetch_b8.
// ---------------------------------------------------------------------------
#if __has_builtin(__builtin_amdgcn_global_load_async_to_lds_b128)
#define HAVE_ASYNC_LDS 1
#else
#define HAVE_ASYNC_LDS 0
#endif

#if HAVE_ASYNC_LDS
typedef int v4i_vs __attribute__((__vector_size__(4 * sizeof(int))));
typedef __attribute__((address_space(1))) v4i_vs* gptr_v4i;
typedef __attribute__((address_space(3))) v4i_vs* lptr_v4i;

__device__ __forceinline__ void async_cp16(const float* g, float4* l) {
    __builtin_amdgcn_global_load_async_to_lds_b128(
        (gptr_v4i)(void*)(g),
        (lptr_v4i)(void*)(l),
        0, 0);
}
__device__ __forceinline__ void wait_async_2() {
#if __has_builtin(__builtin_amdgcn_s_wait_asynccnt)
    __builtin_amdgcn_s_wait_asynccnt(2);
#else
    asm volatile("s_wait_asynccnt 2" ::: "memory");
#endif
}
__device__ __forceinline__ void wait_async_0() {
#if __has_builtin(__builtin_amdgcn_s_wait_asynccnt)
    __builtin_amdgcn_s_wait_asynccnt(0);
#else
    asm volatile("s_wait_asynccnt 0" ::: "memory");
#endif
}
#endif

// ---------------------------------------------------------------------------
// Apply (optional) elu(v*w), scale by coef, and scatter-add one 256-float row.
// Lane owns float4 chunks [lane] and [lane+32] of the row.
// ---------------------------------------------------------------------------
__device__ __forceinline__ void scatter_row(float4 v0, float4 v1,
                                            float4 w0, float4 w1,
                                            int pre, float coef,
                                            float* __restrict__ orow, int lane) {
    if (!pre) {
        v0.x = elu1(v0.x * w0.x); v0.y = elu1(v0.y * w0.y);
        v0.z = elu1(v0.z * w0.z); v0.w = elu1(v0.w * w0.w);
        v1.x = elu1(v1.x * w1.x); v1.y = elu1(v1.y * w1.y);
        v1.z = elu1(v1.z * w1.z); v1.w = elu1(v1.w * w1.w);
    }
    float* a = orow + lane * 4;
    unsafeAtomicAdd(a + 0, v0.x * coef);
    unsafeAtomicAdd(a + 1, v0.y * coef);
    unsafeAtomicAdd(a + 2, v0.z * coef);
    unsafeAtomicAdd(a + 3, v0.w * coef);
    float* b = orow + (lane + 32) * 4;
    unsafeAtomicAdd(b + 0, v1.x * coef);
    unsafeAtomicAdd(b + 1, v1.y * coef);
    unsafeAtomicAdd(b + 2, v1.z * coef);
    unsafeAtomicAdd(b + 3, v1.w * coef);
}

// ---------------------------------------------------------------------------
// Edge scatter kernel, D = 256 specialization. One wave32 per edge,
// grid-stride over edges. Double-buffered async-LDS staging of the gathered
// source row when the gfx1250 async path is available.
// ---------------------------------------------------------------------------
#define EDGE_BLOCK 256
#define WPB (EDGE_BLOCK / 32)

__global__ void __launch_bounds__(EDGE_BLOCK)
edge_scatter_d256(const float* __restrict__ mat, const float* __restrict__ w,
                  const int* __restrict__ src, const int* __restrict__ dst,
                  const int* __restrict__ ef, float* __restrict__ out,
                  int E, int pre) {
    constexpr int D = 256;
    constexpr int D4 = 64;
    const int lane = threadIdx.x & 31;
    const int wib  = (int)(threadIdx.x >> 5);
    const int gw   = (int)((blockIdx.x * (ll)blockDim.x + threadIdx.x) >> 5);
    const int nw   = (int)((gridDim.x * (ll)blockDim.x) >> 5);

    const float4* w4 = (const float4*)w;
    const float4 w0 = w4[lane];
    const float4 w1 = w4[lane + 32];

#if HAVE_ASYNC_LDS
    __shared__ float4 buf[2][WPB][D4];   // 16 KB / block, double buffered

    int e = gw;
    if (e < E) {
        const float* row = mat + (ll)src[e] * D;
        async_cp16(row + lane * 4,        &buf[0][wib][lane]);
        async_cp16(row + (lane + 32) * 4, &buf[0][wib][lane + 32]);
    }
    int p = 0;
    for (; e < E; e += nw) {
        const int en = e + nw;
        if (en < E) {
            const float* row = mat + (ll)src[en] * D;
            async_cp16(row + lane * 4,        &buf[p ^ 1][wib][lane]);
            async_cp16(row + (lane + 32) * 4, &buf[p ^ 1][wib][lane + 32]);
            wait_async_2();   // current buffer's 2 async ops retired
        } else {
            wait_async_0();   // drain: last batch retired
        }
        asm volatile("" ::: "memory");

        const float coef = (ef[e] & 1) ? 1.0f : 2.0f;
        float* orow = out + (ll)dst[e] * D;
        float4 v0 = buf[p][wib][lane];
        float4 v1 = buf[p][wib][lane + 32];
        scatter_row(v0, v1, w0, w1, pre, coef, orow, lane);

        asm volatile("" ::: "memory");
        p ^= 1;
    }
#else
    for (int e = gw; e < E; e += nw) {
        const int en = e + nw;
        if (en < E && lane < 8) {
            // Prefetch next gathered row (8 x 128B lines) -> global_prefetch_b8
            __builtin_prefetch(mat + (ll)src[en] * D + lane * 32, 0, 0);
        }
        const float4* row4 = (const float4*)(mat + (ll)src[e] * D);
        float4 v0 = row4[lane];
        float4 v1 = row4[lane + 32];
        const float coef = (ef[e] & 1) ? 1.0f : 2.0f;
        float* orow = out + (ll)dst[e] * D;
        scatter_row(v0, v1, w0, w1, pre, coef, orow, lane);
    }
#endif
}

// ---------------------------------------------------------------------------
// Generic safety-net kernel (any D). One thread per (edge, column).
// ---------------------------------------------------------------------------
__global__ void edge_scatter_generic(const float* __restrict__ mat, const float* __restrict__ w,
                                     const int* __restrict__ src, const int* __restrict__ dst,
                                     const int* __restrict__ ef, float* __restrict__ out,
                                     ll E, int D, int pre) {
    const ll total  = E * (ll)D;
    const ll stride = gridDim.x * (ll)blockDim.x;
    for (ll i = blockIdx.x * (ll)blockDim.x + threadIdx.x; i < total; i += stride) {
        ll e = i / D;
        int col = (int)(i - e * (ll)D);
        float v = mat[(ll)src[e] * D + col];
        if (!pre) v = elu1(v * w[col]);
        const float coef = (ef[e] & 1) ? 1.0f : 2.0f;
        unsafeAtomicAdd(&out[(ll)dst[e] * D + col], v * coef);
    }
}

// ---------------------------------------------------------------------------
// Host launcher.
// Inputs: [0] graph_embedding (N*D f32), [1] weight (D f32),
//         [2] src (E i32), [3] dst (E i32), [4] e_feat (E i32).
// Output: N*D f32.
// ---------------------------------------------------------------------------
extern "C" void kernel_launch(void* const* d_in, const int* in_sizes, int n_in,
                              void* d_out, int out_size, void* d_ws, size_t ws_size,
                              hipStream_t stream) {
    const float* g   = (const float*)d_in[0];
    const float* w   = (const float*)d_in[1];
    const int*   src = (const int*)d_in[2];
    const int*   dst = (const int*)d_in[3];
    const int*   ef  = (const int*)d_in[4];
    float* out = (float*)d_out;

    const int D  = in_sizes[1];
    const ll  ND = (ll)in_sizes[0];
    const int E  = in_sizes[2];

    // 1) Zero the output.
    if ((ND & 3LL) == 0) {
        ll n4 = ND >> 2;
        int blocks = (int)((n4 + 255) / 256);
        if (blocks < 1) blocks = 1;
        zero_f4<<<blocks, 256, 0, stream>>>((float4*)out, n4);
    } else {
        int blocks = (int)((ND + 255) / 256);
        if (blocks < 1) blocks = 1;
        zero_f1<<<blocks, 256, 0, stream>>>(out, ND);
    }

    // 2) Precompute emb = elu(G*w) into workspace if it fits (avoids ~32x
    //    redundant v_exp per edge). Deterministic: depends only on ws_size.
    int pre = 0;
    if (d_ws != nullptr && (D & 3) == 0 &&
        ws_size >= (size_t)ND * sizeof(float)) {
        pre = 1;
        ll n4 = ND >> 2;
        int blocks = (int)((n4 + 255) / 256);
        if (blocks < 1) blocks = 1;
        emb_elu<<<blocks, 256, 0, stream>>>((const float4*)g, (const float4*)w,
                                            (float4*)d_ws, n4, D >> 2);
    }
    const float* mat = pre ? (const float*)d_ws : g;

    // 3) Edge gather/scale/scatter.
    if (D == 256) {
        // One wave per edge, 8 waves per block; cap grid so each wave loops
        // (enables the async double-buffer pipeline).
        int blocks = (E + WPB - 1) / WPB;
        if (blocks > 16384) blocks = 16384;
        if (blocks < 1) blocks = 1;
        edge_scatter_d256<<<blocks, EDGE_BLOCK, 0, stream>>>(mat, w, src, dst, ef,
                                                             out, E, pre);
    } else {
        ll total = (ll)E * (ll)D;
        ll need  = (total + 255) / 256;
        int blocks = (int)(need < 262144 ? need : 262144);
        if (blocks < 1) blocks = 1;
        edge_scatter_generic<<<blocks, 256, 0, stream>>>(mat, w, src, dst, ef,
                                                         out, (ll)E, D, pre);
    }
}